// Network_7550552507289
// MI455X (gfx1250) — compile-verified
//
#include <hip/hip_runtime.h>
#include <math.h>

typedef __attribute__((ext_vector_type(2))) float v2f;
typedef __attribute__((ext_vector_type(8))) float v8f;

#define B_   128
#define R_   200
#define D1_  32
#define D2_  32
#define D3_  512
#define K1_  100
#define K2_  50
#define EPS_ 1e-5f

__device__ inline v8f wmma_f32(v2f a, v2f b, v8f c) {
  // V_WMMA_F32_16X16X4_F32: D = A(16x4) * B(4x16) + C(16x16), all fp32
  return __builtin_amdgcn_wmma_f32_16x16x4_f32(false, a, false, b, (short)0, c,
                                               false, false);
}

// ---------------- h = relu(pos @ W1)  [M,200]@[200,8] ----------------
__global__ void k_h8(const float* __restrict__ pos, const float* __restrict__ W1,
                     float* __restrict__ h, int M, int K) {
  int idx = blockIdx.x * blockDim.x + threadIdx.x;
  if (idx >= M * 8) return;
  int row = idx >> 3, c = idx & 7;
  const float* p = pos + (size_t)row * K;
  float s = 0.f;
  for (int k = 0; k < K; k++) s += p[k] * W1[k * 8 + c];
  h[idx] = fmaxf(s, 0.f);
}

// ------- fused weight-gen + per-node matvec (WMMA GEMM + epilogue) -------
// T[m, j] = x[m,:] @ Bmat[:, j],  Bmat[i, j<256] = W2[j/32, i*32 + j%32],
//                                 Bmat[i, j>=256] = b2[i*32 + (j-256)]
// xw[m,o] = sum_{c<8} h[m,c]*T[m,c*32+o] + T[m,256+o]
// 18 N-tiles of 16: waves 0..3 each do 4 W2-tiles (t = wave+4i, all < 16);
// waves 0..1 additionally do the bias tile t = 16+wave. The bias predicate is
// forced scalar via readfirstlane so EXEC is untouched around every WMMA.
__global__ void k_weightgen_xw(const float* __restrict__ xin,  // [M,K]
                               const float* __restrict__ W2,   // [8, K*32]
                               const float* __restrict__ b2,   // [K*32]
                               const float* __restrict__ h,    // [M,8]
                               float* __restrict__ xw,         // [M,32]
                               int M, int K) {
  __shared__ float xtile[16][200];
  __shared__ float Ttile[16][288];
  const int tid = threadIdx.x;  // 128 threads = 4 waves
  const int wave = tid >> 5;
  const int lane = tid & 31;
  const int half = lane >> 4;
  const int m = lane & 15;
  const int row0 = blockIdx.x * 16;  // M is a multiple of 16
  const int K32 = K * 32;
  const int waveU = __builtin_amdgcn_readfirstlane(wave);  // scalar wave id
  const bool hasBias = (waveU < 2);                        // scalar branch

  // vectorized LDS fill: rows of xin are 16B aligned (K % 4 == 0)
  const int K4 = K >> 2;
  for (int i = tid; i < 16 * K4; i += 128) {
    int r = i / K4, k4 = i - r * K4;
    const float4 v = ((const float4*)(xin + (size_t)(row0 + r) * K))[k4];
    ((float4*)&xtile[r][0])[k4] = v;
  }
  __syncthreads();

  v8f acc[4], accb;
  for (int i = 0; i < 4; i++)
    for (int q = 0; q < 8; q++) acc[i][q] = 0.f;
  for (int q = 0; q < 8; q++) accb[q] = 0.f;

  for (int k0 = 0; k0 < K; k0 += 4) {
    const int ka = k0 + 2 * half;
    v2f a;
    a.x = xtile[m][ka];
    a.y = xtile[m][ka + 1];
#pragma unroll
    for (int i = 0; i < 4; i++) {
      int t = wave + 4 * i;      // 0..15 -> W2 path, no divergence
      int j = t * 16 + m;
      int c = j >> 5, o = j & 31;
      v2f bb;
      bb.x = W2[c * K32 + ka * 32 + o];
      bb.y = W2[c * K32 + (ka + 1) * 32 + o];
      acc[i] = wmma_f32(a, bb, acc[i]);
    }
    if (hasBias) {               // tile 16+wave -> bias block, scalar branch
      int o = wave * 16 + m;
      v2f bb;
      bb.x = b2[ka * 32 + o];
      bb.y = b2[(ka + 1) * 32 + o];
      accb = wmma_f32(a, bb, accb);
    }
  }
#pragma unroll
  for (int i = 0; i < 4; i++) {
    int t = wave + 4 * i;
#pragma unroll
    for (int v = 0; v < 8; v++) Ttile[v + 8 * half][t * 16 + m] = acc[i][v];
  }
  if (hasBias) {
#pragma unroll
    for (int v = 0; v < 8; v++) Ttile[v + 8 * half][256 + wave * 16 + m] = accb[v];
  }
  __syncthreads();
  for (int i = tid; i < 512; i += 128) {
    int r = i >> 5, o = i & 31;
    size_t gr = row0 + r;
    float s = Ttile[r][256 + o];
    const float* hp = h + gr * 8;
#pragma unroll
    for (int c = 0; c < 8; c++) s += hp[c] * Ttile[r][c * 32 + o];
    xw[gr * 32 + o] = s;
  }
}

// --------------- row softmax of adj with -inf diagonal ---------------
__global__ void k_softmax_attn(const float* __restrict__ adj,
                               float* __restrict__ attn, int rows, int N) {
  int r = blockIdx.x * blockDim.x + threadIdx.x;
  if (r >= rows) return;
  int t = r % N;
  const float* a = adj + (size_t)r * N;
  float* o = attn + (size_t)r * N;
  float mx = -__builtin_inff();
  for (int j = 0; j < N; j++)
    if (j != t) mx = fmaxf(mx, a[j]);
  float s = 0.f;
  for (int j = 0; j < N; j++) {
    float e = (j == t) ? 0.f : expf(a[j] - mx);
    o[j] = e;
    s += e;
  }
  float inv = 1.f / s;
  for (int j = 0; j < N; j++) o[j] *= inv;
}

// --------- message passing: out = attn @ xw + bias (WMMA batched) ---------
__global__ void k_message(const float* __restrict__ attn,  // [B,N,N]
                          const float* __restrict__ xw,    // [B,N,32]
                          const float* __restrict__ bias,  // [32]
                          float* __restrict__ out,         // [B,N,32]
                          int N, int tilesPerB) {
  __shared__ float atile[16][200];
  int blk = blockIdx.x;
  int b = blk / tilesPerB;
  int ti = blk - b * tilesPerB;
  int t0 = ti * 16;
  int tid = threadIdx.x;  // 64 threads = 2 waves
  int wave = tid >> 5, lane = tid & 31, half = lane >> 4, m = lane & 15;

  // vectorized, zero-padded tile fill (rows of attn are 16B aligned)
  const int N4 = N >> 2;
  for (int i = tid; i < 16 * N4; i += 64) {
    int r = i / N4, k4 = i - r * N4;
    int gr = t0 + r;
    int grc = (gr < N) ? gr : (N - 1);                 // clamped (always valid)
    float4 v = ((const float4*)(attn + ((size_t)b * N + grc) * N))[k4];
    if (gr >= N) { v.x = 0.f; v.y = 0.f; v.z = 0.f; v.w = 0.f; }
    ((float4*)&atile[r][0])[k4] = v;
  }
  __syncthreads();

  v8f acc;
  for (int q = 0; q < 8; q++) acc[q] = 0.f;
  int o0 = wave * 16;
  const float* xwb = xw + (size_t)b * N * 32;
  for (int k0 = 0; k0 < N; k0 += 4) {
    int ka = k0 + 2 * half;
    v2f a, bb;
    a.x = atile[m][ka];
    a.y = atile[m][ka + 1];
    bb.x = xwb[ka * 32 + o0 + m];
    bb.y = xwb[(ka + 1) * 32 + o0 + m];
    acc = wmma_f32(a, bb, acc);
  }
#pragma unroll
  for (int v = 0; v < 8; v++) {
    int row = t0 + v + 8 * half;
    if (row < N) {
      int col = o0 + m;
      out[((size_t)b * N + row) * 32 + col] = acc[v] + bias[col];
    }
  }
}

// --------------- TopK pooling: bitonic sort + gathers ---------------
template <int SS>
__global__ void k_pool(const float* __restrict__ xin,     // [B,Nn,32]
                       const float* __restrict__ w,       // [32]
                       const float* __restrict__ adj_in,  // [B,Nn,Nn] or null
                       const float* __restrict__ pos_in,  // [B,Nn,PW] or null
                       float* __restrict__ xout,          // [B,Kk,32]
                       float* __restrict__ adj_out, float* __restrict__ pos_out,
                       float* __restrict__ sig_out,       // [B,Kk] in d_out
                       int Nn, int Kk, int PW) {
  __shared__ float skey[SS];
  __shared__ int sidx[SS];
  int b = blockIdx.x;
  int tid = threadIdx.x;  // blockDim == SS
  float wn = 0.f;
  for (int c = 0; c < 32; c++) wn += w[c] * w[c];
  wn = sqrtf(wn);

  for (int n = tid; n < SS; n += SS) {
    float key = -__builtin_inff();
    if (n < Nn) {
      const float* xr = xin + ((size_t)b * Nn + n) * 32;
      float s = 0.f;
      for (int c = 0; c < 32; c++) s += xr[c] * w[c];
      key = 1.f / (1.f + expf(-s / wn));
    }
    skey[n] = key;
    sidx[n] = n;
  }
  __syncthreads();

  // bitonic sort: descending by score, ascending by index on ties
  for (int k = 2; k <= SS; k <<= 1) {
    for (int j = k >> 1; j > 0; j >>= 1) {
      int i = tid;
      int ixj = i ^ j;
      if (ixj > i) {
        bool up = ((i & k) == 0);
        float s_i = skey[i], s_x = skey[ixj];
        int d_i = sidx[i], d_x = sidx[ixj];
        bool x_first = (s_x > s_i) || (s_x == s_i && d_x < d_i);
        if (x_first == up) {
          skey[i] = s_x; skey[ixj] = s_i;
          sidx[i] = d_x; sidx[ixj] = d_i;
        }
      }
      __syncthreads();
    }
  }

  for (int r = tid; r < Kk; r += SS)
    sig_out[(size_t)b * Kk + r] = 1.f / (1.f + expf(-skey[r]));
  for (int i = tid; i < Kk * 32; i += SS) {
    int r = i >> 5, c = i & 31;
    xout[((size_t)b * Kk + r) * 32 + c] =
        xin[((size_t)b * Nn + sidx[r]) * 32 + c] * skey[r];
  }
  if (adj_out) {
    for (int i = tid; i < Kk * Kk; i += SS) {
      int r = i / Kk, c = i - r * Kk;
      adj_out[((size_t)b * Kk + r) * Kk + c] =
          adj_in[((size_t)b * Nn + sidx[r]) * Nn + sidx[c]];
    }
  }
  if (pos_out) {
    for (int i = tid; i < Kk * PW; i += SS) {
      int r = i / PW, c = i - r * PW;
      pos_out[((size_t)b * Kk + r) * PW + c] =
          pos_in[((size_t)b * Nn + sidx[r]) * PW + c];
    }
  }
}

// ------- augment_adj: a=(adj w/o diag)+I; out=(a@a) w/o diag (WMMA) -------
// Loads are unconditional (clamped addresses); identity/diag/bounds logic is
// applied as selects after the load so EXEC stays all-ones for every WMMA.
__device__ inline float aug_val(const float* __restrict__ a, int i, int j,
                                int N) {
  int ic = (i < N) ? i : (N - 1);
  float raw = a[ic * N + j];           // j < N always at call sites
  float v = (i == j) ? 1.f : raw;
  return (i < N) ? v : 0.f;
}

__global__ void k_augment(const float* __restrict__ adj, float* __restrict__ out,
                          int N, int tN) {
  int blk = blockIdx.x;
  int b = blk / (tN * tN);
  int rem = blk - b * tN * tN;
  int ti = rem / tN, tj = rem - ti * tN;
  int lane = threadIdx.x;  // 32 threads = 1 wave
  int half = lane >> 4, m = lane & 15;
  const float* ab = adj + (size_t)b * N * N;
  int i0 = ti * 16, j0 = tj * 16;
  int gi = i0 + m, gj = j0 + m;
  v8f acc;
  for (int q = 0; q < 8; q++) acc[q] = 0.f;
  for (int k0 = 0; k0 < N; k0 += 4) {
    int ka = k0 + 2 * half;        // ka, ka+1 < N (N % 4 == 0)
    v2f a, bb;
    a.x = aug_val(ab, gi, ka, N);
    a.y = aug_val(ab, gi, ka + 1, N);
    // B element (ka, gj): row index ka always < N; clamp column
    {
      int jc = (gj < N) ? gj : (N - 1);
      float r0 = ab[ka * N + jc];
      float r1 = ab[(ka + 1) * N + jc];
      float v0 = (ka == gj) ? 1.f : r0;
      float v1 = (ka + 1 == gj) ? 1.f : r1;
      bb.x = (gj < N) ? v0 : 0.f;
      bb.y = (gj < N) ? v1 : 0.f;
    }
    acc = wmma_f32(a, bb, acc);
  }
#pragma unroll
  for (int v = 0; v < 8; v++) {
    int ri = i0 + v + 8 * half, rj = j0 + m;
    if (ri < N && rj < N)
      out[(size_t)b * N * N + ri * N + rj] = (ri == rj) ? 0.f : acc[v];
  }
}

// ----------------------- readout [gmp, gap] -----------------------
__global__ void k_readout(const float* __restrict__ xp1,
                          const float* __restrict__ xp2,
                          float* __restrict__ hcat) {
  int b = blockIdx.x, tid = threadIdx.x;  // 64 threads
  if (tid < 32) {
    int o = tid;
    const float* p = xp1 + (size_t)b * K1_ * 32 + o;
    float mx = -__builtin_inff(), sm = 0.f;
    for (int r = 0; r < K1_; r++) { float v = p[r * 32]; mx = fmaxf(mx, v); sm += v; }
    hcat[b * 128 + o] = mx;
    hcat[b * 128 + 32 + o] = sm / (float)K1_;
  } else {
    int o = tid - 32;
    const float* p = xp2 + (size_t)b * K2_ * 32 + o;
    float mx = -__builtin_inff(), sm = 0.f;
    for (int r = 0; r < K2_; r++) { float v = p[r * 32]; mx = fmaxf(mx, v); sm += v; }
    hcat[b * 128 + 64 + o] = mx;
    hcat[b * 128 + 96 + o] = sm / (float)K2_;
  }
}

// ----------------------------- MLP head -----------------------------
__global__ void k_fc_relu(const float* __restrict__ in, const float* __restrict__ W,
                          const float* __restrict__ bias, float* __restrict__ out,
                          int M, int K, int N) {
  int idx = blockIdx.x * blockDim.x + threadIdx.x;
  if (idx >= M * N) return;
  int r = idx / N, c = idx - r * N;
  const float* ir = in + (size_t)r * K;
  float s = bias[c];
  for (int k = 0; k < K; k++) s += ir[k] * W[k * N + c];
  out[idx] = fmaxf(s, 0.f);
}

__global__ void k_bn(const float* __restrict__ in, const float* __restrict__ g,
                     const float* __restrict__ bb, float* __restrict__ out,
                     int M, int N) {
  int c = blockIdx.x * blockDim.x + threadIdx.x;
  if (c >= N) return;
  float s = 0.f, s2 = 0.f;
  for (int r = 0; r < M; r++) { float v = in[r * N + c]; s += v; s2 += v * v; }
  float mean = s / M;
  float var = s2 / M - mean * mean;
  float sc = rsqrtf(var + EPS_) * g[c];
  float sh = bb[c];
  for (int r = 0; r < M; r++) out[r * N + c] = (in[r * N + c] - mean) * sc + sh;
}

__global__ void k_fc3_logsoftmax(const float* __restrict__ in,
                                 const float* __restrict__ W,
                                 const float* __restrict__ bias,
                                 float* __restrict__ out) {
  int r = threadIdx.x;  // 128
  const float* ir = in + (size_t)r * D3_;
  float l0 = bias[0], l1 = bias[1];
  for (int k = 0; k < D3_; k++) { float v = ir[k]; l0 += v * W[k * 2]; l1 += v * W[k * 2 + 1]; }
  float mx = fmaxf(l0, l1);
  float lse = mx + logf(expf(l0 - mx) + expf(l1 - mx));
  out[r * 2] = l0 - lse;
  out[r * 2 + 1] = l1 - lse;
}

__global__ void k_copyw(const float* __restrict__ w1, const float* __restrict__ w2,
                        float* __restrict__ out) {
  int t = threadIdx.x;
  if (t < 32) out[256 + t] = w1[t];
  else if (t < 64) out[288 + (t - 32)] = w2[t - 32];
}

// ----------------------------- launcher -----------------------------
extern "C" void kernel_launch(void* const* d_in, const int* in_sizes, int n_in,
                              void* d_out, int out_size, void* d_ws, size_t ws_size,
                              hipStream_t stream) {
  const float* x       = (const float*)d_in[0];   // [128,200,200]
  const float* adj     = (const float*)d_in[1];   // [128,200,200]
  const float* pos     = (const float*)d_in[2];   // [128,200,200]
  const float* n1_W1   = (const float*)d_in[3];
  const float* n1_W2   = (const float*)d_in[4];
  const float* n1_b2   = (const float*)d_in[5];
  const float* conv1_b = (const float*)d_in[6];
  const float* pool1_w = (const float*)d_in[7];
  const float* n2_W1   = (const float*)d_in[8];
  const float* n2_W2   = (const float*)d_in[9];
  const float* n2_b2   = (const float*)d_in[10];
  const float* conv2_b = (const float*)d_in[11];
  const float* pool2_w = (const float*)d_in[12];
  const float* fc1_W   = (const float*)d_in[13];
  const float* fc1_b   = (const float*)d_in[14];
  const float* bn1_g   = (const float*)d_in[15];
  const float* bn1_b   = (const float*)d_in[16];
  const float* fc2_W   = (const float*)d_in[17];
  const float* fc2_b   = (const float*)d_in[18];
  const float* bn2_g   = (const float*)d_in[19];
  const float* bn2_b   = (const float*)d_in[20];
  const float* fc3_W   = (const float*)d_in[21];
  const float* fc3_b   = (const float*)d_in[22];
  float* dout = (float*)d_out;

  const int M1 = B_ * R_;   // 25600
  const int M2 = B_ * K1_;  // 12800

  float* ws = (float*)d_ws;
  size_t off = 0;
  auto alloc = [&](size_t n) { float* p = ws + off; off += n; return p; };
  float* h1    = alloc((size_t)M1 * 8);
  float* xw1   = alloc((size_t)M1 * 32);
  float* attn1 = alloc((size_t)B_ * R_ * R_);
  float* xc1   = alloc((size_t)M1 * 32);
  float* xp1   = alloc((size_t)B_ * K1_ * 32);
  float* adjp1 = alloc((size_t)B_ * K1_ * K1_);
  float* posp1 = alloc((size_t)B_ * K1_ * R_);
  float* adj2  = alloc((size_t)B_ * K1_ * K1_);
  float* h2    = alloc((size_t)M2 * 8);
  float* xw2   = alloc((size_t)M2 * 32);
  float* attn2 = alloc((size_t)B_ * K1_ * K1_);
  float* xc2   = alloc((size_t)M2 * 32);
  float* xp2   = alloc((size_t)B_ * K2_ * 32);
  float* hcat  = alloc((size_t)B_ * 128);
  float* f1    = alloc((size_t)B_ * 32);
  float* f1b   = alloc((size_t)B_ * 32);
  float* f2    = alloc((size_t)B_ * 512);
  float* f2b   = alloc((size_t)B_ * 512);
  (void)off; (void)ws_size; (void)in_sizes; (void)n_in; (void)out_size;

  // ---- conv1 ----
  k_h8<<<(M1 * 8 + 255) / 256, 256, 0, stream>>>(pos, n1_W1, h1, M1, R_);
  k_weightgen_xw<<<M1 / 16, 128, 0, stream>>>(x, n1_W2, n1_b2, h1, xw1, M1, R_);
  k_softmax_attn<<<(B_ * R_ + 255) / 256, 256, 0, stream>>>(adj, attn1, B_ * R_, R_);
  k_message<<<13 * B_, 64, 0, stream>>>(attn1, xw1, conv1_b, xc1, R_, 13);
  // ---- pool1 (writes sigmoid(s1) straight into d_out[320..13120)) ----
  k_pool<256><<<B_, 256, 0, stream>>>(xc1, pool1_w, adj, pos, xp1, adjp1, posp1,
                                      dout + 320, R_, K1_, R_);
  // ---- augment_adj ----
  k_augment<<<B_ * 49, 32, 0, stream>>>(adjp1, adj2, K1_, 7);
  // ---- conv2 ----
  k_h8<<<(M2 * 8 + 255) / 256, 256, 0, stream>>>(posp1, n2_W1, h2, M2, R_);
  k_weightgen_xw<<<M2 / 16, 128, 0, stream>>>(xp1, n2_W2, n2_b2, h2, xw2, M2, 32);
  k_softmax_attn<<<(B_ * K1_ + 255) / 256, 256, 0, stream>>>(adj2, attn2, B_ * K1_, K1_);
  k_message<<<7 * B_, 64, 0, stream>>>(attn2, xw2, conv2_b, xc2, K1_, 7);
  // ---- pool2 (writes sigmoid(s2) into d_out[13120..19520)) ----
  k_pool<128><<<B_, 128, 0, stream>>>(xc2, pool2_w, nullptr, nullptr, xp2, nullptr,
                                      nullptr, dout + 13120, K1_, K2_, 0);
  // ---- readout + head ----
  k_readout<<<B_, 64, 0, stream>>>(xp1, xp2, hcat);
  k_fc_relu<<<(B_ * 32 + 255) / 256, 256, 0, stream>>>(hcat, fc1_W, fc1_b, f1, B_, 128, 32);
  k_bn<<<1, 32, 0, stream>>>(f1, bn1_g, bn1_b, f1b, B_, 32);
  k_fc_relu<<<(B_ * 512 + 255) / 256, 256, 0, stream>>>(f1b, fc2_W, fc2_b, f2, B_, 32, 512);
  k_bn<<<1, 512, 0, stream>>>(f2, bn2_g, bn2_b, f2b, B_, 512);
  k_fc3_logsoftmax<<<1, 128, 0, stream>>>(f2b, fc3_W, fc3_b, dout);
  k_copyw<<<1, 64, 0, stream>>>(pool1_w, pool2_w, dout);
}